// LGA2_15796889715119
// MI455X (gfx1250) — compile-verified
//
#include <hip/hip_runtime.h>
#include <stdint.h>

// Problem constants (fixed by the reference setup_inputs()).
#define B_   2
#define D_   48
#define H_   320
#define W_   640
#define R_   2
#define K_   5
#define K2_  25
#define C_   75           // 3 * k * k guidance channels
#define WT   128          // pixels (threads) per block along W
#define RP   (WT + 2*R_)  // 132: staged tile row pitch in floats
#define PLANE_F (K_ * RP) // 660 floats per staged plane tile
#define NSLOT 8           // ring slots (power of two: slot = p & 7)
#define SMASK (NSLOT - 1)
#define ZSLOT NSLOT       // extra slot kept at zero for d-1<0 / d+1>=D

#if __has_builtin(__builtin_amdgcn_tensor_load_to_lds) && __has_builtin(__builtin_amdgcn_s_wait_tensorcnt)
#define HAVE_TDM 1
#else
#define HAVE_TDM 0
#endif

#if HAVE_TDM
typedef unsigned int u32x4 __attribute__((ext_vector_type(4)));
typedef int          i32x4 __attribute__((ext_vector_type(4)));
typedef int          i32x8 __attribute__((ext_vector_type(8)));

// Issue one TDM 2-D tile load: 5 rows x 132 floats, global row stride W_,
// into LDS at `ldst`. Descriptor per CDNA5 ISA.
// Tile dims == tensor dims -> plain strided copy, no OOB zeroing needed
// (only called for fully-interior tiles).
__device__ __forceinline__ void tdm_load_tile(const float* gsrc, const float* ldst)
{
    uint64_t ga = (uint64_t)(uintptr_t)gsrc;       // tile start VA (57-bit field)
    uint32_t la = (uint32_t)(uintptr_t)ldst;       // flat LDS addr -> low 32 = LDS byte offset
    u32x4 g0;
    g0[0] = 1u;                                    // count=1 (valid), not restore, no gather
    g0[1] = la;                                    // lds_addr [63:32]
    g0[2] = (uint32_t)ga;                          // global_addr [95:64]
    g0[3] = ((uint32_t)(ga >> 32) & 0x01FFFFFFu)   // global_addr [120:96]
          | (2u << 30);                            // type=2 ("image") [127:126]
    i32x8 g1;
    g1[0] = (int)(2u << 16);                       // workgroup_mask=0, data_size=2 (4B)
    g1[1] = (int)((uint32_t)RP << 16);             // tensor_dim0[15:0] in [63:48]
    g1[2] = (int)(((uint32_t)RP >> 16) | (5u << 16)); // tensor_dim0[31:16] | tensor_dim1[15:0]
    g1[3] = (int)((uint32_t)RP << 16);             // tensor_dim1[31:16]=0 | tile_dim0=132
    g1[4] = 5;                                     // tile_dim1=5 | tile_dim2=0
    g1[5] = W_;                                    // tensor_dim0_stride[31:0] = 640 elems
    g1[6] = 0;                                     // stride hi | tensor_dim1_stride lo
    g1[7] = 0;
    i32x4 z4 = {0, 0, 0, 0};
#if defined(__clang_major__) && (__clang_major__ >= 23)
    i32x8 z8 = {0, 0, 0, 0, 0, 0, 0, 0};
    __builtin_amdgcn_tensor_load_to_lds(g0, g1, z4, z4, z8, 0);
#else
    __builtin_amdgcn_tensor_load_to_lds(g0, g1, z4, z4, 0);
#endif
}
#endif // HAVE_TDM

// Cooperative (all 128 threads) fallback loader with zero padding, for blocks
// whose halo leaves the image (and for toolchains without the TDM builtin).
__device__ __forceinline__ void load_plane_manual(float* dst, const float* xb,
                                                  int p, int h, int w0)
{
    const float* src = xb + (size_t)p * H_ * W_;
#pragma unroll
    for (int i = 0; i < K_; ++i) {
        const int hh = h - R_ + i;
        const bool rowok = (hh >= 0) && (hh < H_);
#pragma unroll
        for (int c0 = 0; c0 < RP; c0 += WT) {
            const int c = c0 + (int)threadIdx.x;
            if (c < RP) {
                const int ww = w0 - R_ + c;
                float v = 0.0f;
                if (rowok && (ww >= 0) && (ww < W_)) v = src[(size_t)hh * W_ + ww];
                dst[i * RP + c] = v;
            }
        }
    }
}

__global__ __launch_bounds__(WT)
void lga_pass_kernel(const float* __restrict__ x,
                     const float* __restrict__ wg,
                     float* __restrict__ out)
{
    __shared__ float smem[(NSLOT + 1) * PLANE_F];   // 8 ring slots + zero slot

    const int tid  = (int)threadIdx.x;
    const int w0   = (int)blockIdx.x * WT;
    const int h    = (int)blockIdx.y;
    const int b    = (int)blockIdx.z;
    const int wpix = w0 + tid;

    // ---- per-pixel guidance weights into VGPRs (reused across all 48 d) ----
    float wt[C_];
    {
        const size_t cstride = (size_t)H_ * W_;
        const float* wp = wg + (size_t)b * C_ * cstride + (size_t)h * W_ + wpix;
#pragma unroll
        for (int c = 0; c < C_; ++c) wt[c] = wp[(size_t)c * cstride];
    }

    const float* xb = x + (size_t)b * D_ * H_ * W_;
    const float* gt = xb + (size_t)(h - R_) * W_ + (w0 - R_);  // halo tile base (interior only)
    float* ob = out + (size_t)b * D_ * H_ * W_ + (size_t)h * W_ + wpix;

    const bool interior = (h >= R_) && (h <= H_ - 1 - R_) &&
                          (w0 >= R_) && (w0 + WT - 1 + R_ <= W_ - 1);
#if HAVE_TDM
    const bool use_tdm = interior;
#else
    const bool use_tdm = false; (void)interior; (void)gt;
#endif

    // zero slot (read for d-1 at d==0 and d+1 at d==47)
    for (int i = tid; i < PLANE_F; i += WT) smem[ZSLOT * PLANE_F + i] = 0.0f;

    // ---- preload planes 0..2 ----
    if (use_tdm) {
#if HAVE_TDM
        if (tid == 0) {
#pragma unroll
            for (int p = 0; p < 3; ++p)
                tdm_load_tile(gt + (size_t)p * H_ * W_, &smem[(p & SMASK) * PLANE_F]);
        }
#endif
    } else {
        for (int p = 0; p < 3; ++p)
            load_plane_manual(&smem[(p & SMASK) * PLANE_F], xb, p, h, w0);
    }

    // ---- main loop: two disparities per iteration, prefetch two planes ----
    for (int t = 0; t < D_ / 2; ++t) {
        const int d = 2 * t;
        __syncthreads();                 // everyone done reading slots we overwrite
        const int p3 = d + 3, p4 = d + 4;
        if (use_tdm) {
#if HAVE_TDM
            if (tid == 0) {
                if (p3 < D_) tdm_load_tile(gt + (size_t)p3 * H_ * W_, &smem[(p3 & SMASK) * PLANE_F]);
                if (p4 < D_) tdm_load_tile(gt + (size_t)p4 * H_ * W_, &smem[(p4 & SMASK) * PLANE_F]);
                // planes <= d+2 must be complete; allow the just-issued ones in flight
                if      (p4 < D_) __builtin_amdgcn_s_wait_tensorcnt(2);
                else if (p3 < D_) __builtin_amdgcn_s_wait_tensorcnt(1);
                else              __builtin_amdgcn_s_wait_tensorcnt(0);
            }
#endif
        } else {
            if (p3 < D_) load_plane_manual(&smem[(p3 & SMASK) * PLANE_F], xb, p3, h, w0);
            if (p4 < D_) load_plane_manual(&smem[(p4 & SMASK) * PLANE_F], xb, p4, h, w0);
        }
        __syncthreads();                 // staged data visible to all waves

        const float* Pm = (d == 0)     ? &smem[ZSLOT * PLANE_F] : &smem[((d - 1) & SMASK) * PLANE_F];
        const float* P0 = &smem[(d & SMASK) * PLANE_F];
        const float* P1 = &smem[((d + 1) & SMASK) * PLANE_F];
        const float* P2 = (d + 2 < D_) ? &smem[((d + 2) & SMASK) * PLANE_F] : &smem[ZSLOT * PLANE_F];

        float acc0 = 0.0f, acc1 = 0.0f;
#pragma unroll
        for (int i = 0; i < K_; ++i) {
            const int base = i * RP + tid;
            float am[K_], a0[K_], a1[K_], a2[K_];
#pragma unroll
            for (int j = 0; j < K_; ++j) {
                am[j] = Pm[base + j];
                a0[j] = P0[base + j];
                a1[j] = P1[base + j];
                a2[j] = P2[base + j];
            }
#pragma unroll
            for (int j = 0; j < K_; ++j) {
                const int c = i * K_ + j;
                acc0 = fmaf(wt[c],           a0[j], acc0);   // w0 * x[d]
                acc0 = fmaf(wt[K2_ + c],     am[j], acc0);   // w1 * x[d-1]
                acc0 = fmaf(wt[2 * K2_ + c], a1[j], acc0);   // w2 * x[d+1]
                acc1 = fmaf(wt[c],           a1[j], acc1);   // w0 * x[d+1]
                acc1 = fmaf(wt[K2_ + c],     a0[j], acc1);   // w1 * x[d]
                acc1 = fmaf(wt[2 * K2_ + c], a2[j], acc1);   // w2 * x[d+2]
            }
        }
        ob[(size_t)d * H_ * W_]       = acc0;
        ob[(size_t)(d + 1) * H_ * W_] = acc1;
    }
}

extern "C" void kernel_launch(void* const* d_in, const int* in_sizes, int n_in,
                              void* d_out, int out_size, void* d_ws, size_t ws_size,
                              hipStream_t stream)
{
    (void)in_sizes; (void)n_in; (void)out_size; (void)ws_size;
    const float* x = (const float*)d_in[0];
    const float* w = (const float*)d_in[1];
    float* y   = (float*)d_ws;    // intermediate y = LGA(x, w): 78.6 MB
    float* out = (float*)d_out;

    dim3 grid(W_ / WT, H_, B_);   // 5 x 320 x 2 blocks
    dim3 block(WT);               // 128 threads = 4 wave32
    lga_pass_kernel<<<grid, block, 0, stream>>>(x, w, y);
    lga_pass_kernel<<<grid, block, 0, stream>>>(y, w, out);
}